// NeighborSelector_60868276519654
// MI455X (gfx1250) — compile-verified
//
#include <hip/hip_runtime.h>
#include <cstdint>

// ---------------------------------------------------------------------------
// NeighborSelector for MI455X (gfx1250).
// Roofline: 512 MB candidate stream / 23.3 TB/s ~= 22 us floor -> memory bound.
// dist_kernel = wave-per-row streaming reduction fed by a depth-3 async
// global->LDS pipeline (ASYNCcnt). head kernel = the reference's one real
// matmul, done on V_WMMA_F32_16X16X4_F32.
// ---------------------------------------------------------------------------

constexpr int BB = 32;
constexpr int NN = 8192;
constexpr int DD = 512;     // floats per row; 128 float4
constexpr int CHUNKS = 32;  // chunks per batch for streaming kernel
constexpr int WAVES = 8;    // 256 threads/block, wave32
constexpr int CAND_PER_BLOCK = NN / CHUNKS;          // 256
constexpr int CAND_PER_WAVE = CAND_PER_BLOCK / WAVES; // 32

typedef __attribute__((ext_vector_type(2))) float v2f;
typedef __attribute__((ext_vector_type(8))) float v8f;
typedef int gv4i __attribute__((vector_size(16)));  // matches builtin param type

#define AS1 __attribute__((address_space(1)))
#define AS3 __attribute__((address_space(3)))

#if __has_builtin(__builtin_amdgcn_global_load_async_to_lds_b128)
#define HAS_ASYNC 1
#else
#define HAS_ASYNC 0
#endif

#if HAS_ASYNC
__device__ inline AS1 gv4i* to_global(const void* p) {
  return (AS1 gv4i*)p;  // generic -> global addrspacecast (+ drop const)
}
__device__ inline AS3 gv4i* to_lds(void* p) {
  return (AS3 gv4i*)p;  // generic -> LDS addrspacecast
}
#endif

template <int Cnt>
__device__ inline void wait_async() {
#if __has_builtin(__builtin_amdgcn_s_wait_asynccnt)
  __builtin_amdgcn_s_wait_asynccnt(Cnt);
#else
  asm volatile("s_wait_asynccnt %0" ::"i"(Cnt) : "memory");
#endif
  asm volatile("" ::: "memory");
}

// ---------------------------------------------------------------------------
// Kernel 1: score head  base(32x6) = [metric;transport](32x1024) @ W^T + b
// via V_WMMA_F32_16X16X4_F32 (two 16-row tiles, one per wave).
// Branchless operand feeds: K-loop split into metric/transport phases;
// W rows clamped + lane-masked so no exec-mask diamonds around the WMMAs.
// ---------------------------------------------------------------------------
__global__ __launch_bounds__(64) void head_wmma(const float* __restrict__ metric,
                                                const float* __restrict__ transport,
                                                const float* __restrict__ W,
                                                const float* __restrict__ bias,
                                                float* __restrict__ base /*32x8*/) {
#if __has_builtin(__builtin_amdgcn_wmma_f32_16x16x4_f32)
  const int lane = threadIdx.x & 31;
  const int wave = threadIdx.x >> 5;  // tile: batches [wave*16, wave*16+16)
  const int rowBase = wave * 16;
  const int half = lane >> 4;   // 0: lanes 0-15 hold K%4 in {0,1}; 1: {2,3}
  const int lid = lane & 15;

  // A row (per lane): batch row of metric / transport.
  const float* am = metric + (size_t)(rowBase + lid) * DD;
  const float* at = transport + (size_t)(rowBase + lid) * DD;
  // B col (per lane): W row, clamped to a valid row; masked to 0 for lid>=6.
  const int wrow = (lid < 6) ? lid : 5;
  const float wmask = (lid < 6) ? 1.0f : 0.0f;
  const float* wr = W + (size_t)wrow * (2 * DD);

  v8f acc = {};
#pragma unroll 1
  for (int phase = 0; phase < 2; ++phase) {
    const float* ap = phase ? at : am;
    const int kwbase = phase * DD;
    for (int k0 = 0; k0 < DD; k0 += 4) {
      const int ka = k0 + half * 2;  // this lane's K pair within the 4-step
      v2f a, bv;
      a.x = ap[ka];
      a.y = ap[ka + 1];
      bv.x = wr[kwbase + ka] * wmask;
      bv.y = wr[kwbase + ka + 1] * wmask;
      acc = __builtin_amdgcn_wmma_f32_16x16x4_f32(false, a, false, bv, (short)0,
                                                  acc, false, false);
    }
  }
  // C/D 16x16 f32 layout: lanes 0-15: vgpr r = C[M=r][N=lane];
  //                       lanes 16-31: vgpr r = C[M=8+r][N=lane-16]
  const int mrow = rowBase + half * 8;
  if (lid < 6) {
    const float bj = bias[lid];
#pragma unroll
    for (int r = 0; r < 8; ++r) base[(mrow + r) * 8 + lid] = acc[r] + bj;
  }
#else
  for (int e = threadIdx.x; e < BB * 6; e += blockDim.x) {
    const int row = e / 6, j = e % 6;
    float s = bias[j];
    for (int k = 0; k < DD; ++k) s += metric[row * DD + k] * W[j * (2 * DD) + k];
    for (int k = 0; k < DD; ++k)
      s += transport[row * DD + k] * W[j * (2 * DD) + DD + k];
    base[row * 8 + j] = s;
  }
#endif
}

// ---------------------------------------------------------------------------
// Kernel 2: streaming distances.  One wave per candidate row (512 f32),
// depth-3 async global->LDS prefetch pipeline (4 LDS buffers per wave).
// ---------------------------------------------------------------------------
__global__ __launch_bounds__(256) void dist_kernel(const float* __restrict__ query,
                                                   const float* __restrict__ cand,
                                                   const float* __restrict__ metric,
                                                   const float* __restrict__ transport,
                                                   float* __restrict__ dist,
                                                   float* __restrict__ corr) {
  const int lane = threadIdx.x & 31;
  const int wave = threadIdx.x >> 5;
  const int batch = blockIdx.x / CHUNKS;
  const int chunk = blockIdx.x % CHUNKS;
  const int n0 = chunk * CAND_PER_BLOCK + wave * CAND_PER_WAVE;

  // Per-lane slices of query / metric / transport (reused for every row).
  const float4* q4p = (const float4*)(query + (size_t)batch * DD);
  const float4* m4p = (const float4*)(metric + (size_t)batch * DD);
  const float4* t4p = (const float4*)(transport + (size_t)batch * DD);
  float4 q4[4], m4[4], t4[4];
#pragma unroll
  for (int s = 0; s < 4; ++s) {
    q4[s] = q4p[lane + 32 * s];
    m4[s] = m4p[lane + 32 * s];
    t4[s] = t4p[lane + 32 * s];
  }

  const float* rowbase = cand + ((size_t)batch * NN + n0) * DD;

  auto consume = [&](const float4* c4p, int i) {
    float accd = 0.0f, acct = 0.0f;
#pragma unroll
    for (int s = 0; s < 4; ++s) {
      const float4 c = c4p[lane + 32 * s];
      const float4 q = q4[s], m = m4[s], t = t4[s];
      float d;
      d = c.x - q.x; accd = fmaf(d * m.x, d, accd); acct = fmaf(d, t.x, acct);
      d = c.y - q.y; accd = fmaf(d * m.y, d, accd); acct = fmaf(d, t.y, acct);
      d = c.z - q.z; accd = fmaf(d * m.z, d, accd); acct = fmaf(d, t.z, acct);
      d = c.w - q.w; accd = fmaf(d * m.w, d, accd); acct = fmaf(d, t.w, acct);
    }
#pragma unroll
    for (int off = 16; off > 0; off >>= 1) {
      accd += __shfl_xor(accd, off, 32);
      acct += __shfl_xor(acct, off, 32);
    }
    if (lane == 0) {
      const float dd = sqrtf(fmaxf(accd, 1e-8f));
      const size_t o = (size_t)batch * NN + n0 + i;
      dist[o] = dd;
      corr[o] = dd + 0.1f * acct;
    }
  };

#if HAS_ASYNC
  __shared__ float lbuf[WAVES][4][DD];  // 64 KB: 4 row buffers per wave
  float* bufs = &lbuf[wave][0][0];
  auto stage = [&](int buf, int i) {
    // IOFFSET is added to BOTH the global and the LDS address (ISA 15.18.3),
    // so one per-lane base covers all four 512B segments of the 2KB row.
    const char* g = (const char*)(rowbase + (size_t)i * DD) + lane * 16;
    char* l = (char*)(bufs + buf * DD) + lane * 16;
    __builtin_amdgcn_global_load_async_to_lds_b128(to_global(g), to_lds(l), 0, 0);
    __builtin_amdgcn_global_load_async_to_lds_b128(to_global(g), to_lds(l), 512, 0);
    __builtin_amdgcn_global_load_async_to_lds_b128(to_global(g), to_lds(l), 1024, 0);
    __builtin_amdgcn_global_load_async_to_lds_b128(to_global(g), to_lds(l), 1536, 0);
  };
  stage(0, 0);
  stage(1, 1);
  stage(2, 2);
  for (int i = 0; i < CAND_PER_WAVE; ++i) {
    // Async loads complete in order: after issuing stage(i+3), waiting for
    // asynccnt<=12 guarantees stage(i)'s 4 transfers have landed.
    if (i + 3 < CAND_PER_WAVE) { stage((i + 3) & 3, i + 3); wait_async<12>(); }
    else if (i + 2 < CAND_PER_WAVE) wait_async<8>();
    else if (i + 1 < CAND_PER_WAVE) wait_async<4>();
    else wait_async<0>();
    consume((const float4*)(bufs + (i & 3) * DD), i);
  }
#else
  for (int i = 0; i < CAND_PER_WAVE; ++i)
    consume((const float4*)(rowbase + (size_t)i * DD), i);
#endif
}

// ---------------------------------------------------------------------------
// Kernel 3: per-batch top-32 by distance (stable ties like jax.lax.top_k),
// fixed attractor/repulsor indices 0..15 (their channels are candidate-
// invariant, and lax.top_k breaks ties toward low indices), gather
// embeddings, emit scores.
// ---------------------------------------------------------------------------
__global__ __launch_bounds__(256) void select_kernel(const float* __restrict__ cand,
                                                     const float* __restrict__ dist,
                                                     const float* __restrict__ corr,
                                                     const float* __restrict__ base,
                                                     float* __restrict__ out_emb,
                                                     float* __restrict__ out_scores,
                                                     float* __restrict__ out_idx) {
  const int b = blockIdx.x;
  const int t = threadIdx.x;
  __shared__ float sdist[NN];   // 32 KB
  __shared__ float rval[256];
  __shared__ int ridx[256];
  __shared__ int sel[64];

  for (int j = t; j < NN; j += 256) sdist[j] = dist[(size_t)b * NN + j];
  __syncthreads();

  for (int it = 0; it < 32; ++it) {
    float best = 3.4e38f;
    int bidx = 0x7fffffff;
    for (int j = t; j < NN; j += 256) {
      const float v = sdist[j];
      if (v < best) { best = v; bidx = j; }  // strict '<' keeps lowest index
    }
    rval[t] = best; ridx[t] = bidx;
    __syncthreads();
#pragma unroll
    for (int s = 128; s > 0; s >>= 1) {
      if (t < s) {
        const float v2 = rval[t + s]; const int i2 = ridx[t + s];
        if (v2 < rval[t] || (v2 == rval[t] && i2 < ridx[t])) {
          rval[t] = v2; ridx[t] = i2;
        }
      }
      __syncthreads();
    }
    if (t == 0) { sel[it] = ridx[0]; sdist[ridx[0]] = 3.4e38f; }
    __syncthreads();
  }
  if (t >= 32 && t < 48) sel[t] = t - 32;       // attractors: ties -> 0..15
  else if (t >= 48 && t < 64) sel[t] = t - 48;  // repulsors:  ties -> 0..15
  __syncthreads();

  // scores + indices
  if (t < 64) {
    const int idx = sel[t];
    const float dv = dist[(size_t)b * NN + idx];
    const float cv = corr[(size_t)b * NN + idx];
    float* sc = out_scores + ((size_t)b * 64 + t) * 6;
    sc[0] = 1.0f / (dv + 0.001f);
    sc[1] = 1.0f / (fabsf(cv) + 0.001f);
    sc[2] = base[b * 8 + 2];
    sc[3] = base[b * 8 + 3];
    sc[4] = base[b * 8 + 4];
    sc[5] = (float)t / 64.0f;
    out_idx[(size_t)b * 64 + t] = (float)idx;
  }

  // embeddings gather: 64 rows x 128 float4 per batch
  for (int e = t; e < 64 * 128; e += 256) {
    const int s = e >> 7, d4 = e & 127;
    const float4* src = (const float4*)(cand + ((size_t)b * NN + sel[s]) * DD);
    ((float4*)out_emb)[((size_t)b * 64 + s) * 128 + d4] = src[d4];
  }
}

// ---------------------------------------------------------------------------
extern "C" void kernel_launch(void* const* d_in, const int* in_sizes, int n_in,
                              void* d_out, int out_size, void* d_ws, size_t ws_size,
                              hipStream_t stream) {
  const float* query     = (const float*)d_in[0];
  const float* cand      = (const float*)d_in[1];
  const float* metric    = (const float*)d_in[2];
  const float* transport = (const float*)d_in[3];
  const float* W         = (const float*)d_in[4];
  const float* bias      = (const float*)d_in[5];

  float* ws   = (float*)d_ws;
  float* dist = ws;                       // B*N
  float* corr = ws + (size_t)BB * NN;     // B*N
  float* base = ws + (size_t)2 * BB * NN; // B*8

  float* out_emb    = (float*)d_out;                       // 32*64*512
  float* out_scores = out_emb + (size_t)BB * 64 * DD;      // 32*64*6
  float* out_idx    = out_scores + (size_t)BB * 64 * 6;    // 32*64 (as float)

  head_wmma<<<1, 64, 0, stream>>>(metric, transport, W, bias, base);
  dist_kernel<<<BB * CHUNKS, 256, 0, stream>>>(query, cand, metric, transport,
                                               dist, corr);
  select_kernel<<<BB, 256, 0, stream>>>(cand, dist, corr, base, out_emb,
                                        out_scores, out_idx);
}